// HeightMapDenoiseLoss_15839839388110
// MI455X (gfx1250) — compile-verified
//
#include <hip/hip_runtime.h>
#include <math.h>

// CDNA5 / gfx1250: wave32, WMMA 16x16x4 f32.
typedef __attribute__((ext_vector_type(2))) float v2f;
typedef __attribute__((ext_vector_type(8))) float v8f;

#define XS     540
#define YS     540
#define NPIX   (XS * YS)      // 291600 pixels per batch sample
#define NB     64             // boxes per sample
#define BATCH  4
#define CHUNKS 64             // blocks per batch (fixed -> deterministic partials)

// Rasterize boxes via WMMA (affine transform as 16x16x4 f32 GEMM) and
// accumulate the masked BCE+focal partial sums per block.
__global__ __launch_bounds__(256) void raster_loss_kernel(
    const float* __restrict__ logits, const float* __restrict__ boxes,
    const float* __restrict__ hmaps, float* __restrict__ partial)
{
    // Per-box derived params: {c, s, t0, t1, w/2, l/2, hv, pad}
    __shared__ float sbox[NB][8];
    __shared__ float sred[8][4];

    const int b   = blockIdx.y;
    const int tid = threadIdx.x;

    if (tid < NB) {
        const float* bp = boxes + (size_t)(b * NB + tid) * 7;
        float cx = bp[0], cy = bp[1], w = bp[3], l = bp[4], h = bp[5], th = bp[6];
        float cxg = (cx + 54.f) * 5.f;      // (cx - PC_RANGE[0]) / GRID
        float cyg = (cy + 54.f) * 5.f;
        float c = cosf(th);                 // cos(-theta)
        float s = -sinf(th);                // sin(-theta)
        sbox[tid][0] = c;
        sbox[tid][1] = s;
        sbox[tid][2] = -(cxg * c - cyg * s);   // t0: lx = x*c - y*s + t0
        sbox[tid][3] = -(cxg * s + cyg * c);   // t1: ly = x*s + y*c + t1
        sbox[tid][4] = w * 5.f * 0.5f;         // half width in grid units
        sbox[tid][5] = l * 5.f * 0.5f;         // half length in grid units
        sbox[tid][6] = h * 0.2f;               // hv = h / (PC_RANGE[5] + 2)
        sbox[tid][7] = 0.f;
    }
    __syncthreads();

    const int  lane = tid & 31;
    const int  wave = tid >> 5;
    const int  lo   = lane & 15;
    const bool hi   = lane >= 16;
    const int  hi4  = hi ? 4 : 0;

    // A fragments for 8 groups of 8 boxes. 16x4 f32 A layout: lane L%16 = row M,
    // low lanes hold K=0,1 (rotation), high lanes K=2,3 (translation, 0).
    v2f af[8];
#pragma unroll
    for (int g = 0; g < 8; ++g) {
        int j = g * 8 + (lo >> 1);
        float c = sbox[j][0], s = sbox[j][1], t0 = sbox[j][2], t1 = sbox[j][3];
        v2f a;
        if (!hi) { a.x = (lane & 1) ? s  : c;  a.y = (lane & 1) ? c : -s; }
        else     { a.x = (lane & 1) ? t1 : t0; a.y = 0.f; }
        af[g] = a;
    }

    float acc_b = 0.f, acc_f = 0.f, acc_c = 0.f;

    const int stride = CHUNKS * 256;  // pixels covered per sweep (64 blocks * 8 waves * 32)
    for (int base = blockIdx.x * 256 + wave * 32; base < NPIX; base += stride) {
        // Two 16-pixel tiles: tile0 = base..base+15, tile1 = base+16..base+31.
        // B 4x16 layout: lane = column (pixel), low lanes [x,y], high lanes [1,0].
        int p0 = base + lo;
        int p1 = base + 16 + lo;
        v2f b0, b1;
        if (!hi) {
            b0.x = (float)(p0 % XS); b0.y = (float)(p0 / XS);
            b1.x = (float)(p1 % XS); b1.y = (float)(p1 / XS);
        } else {
            b0.x = 1.f; b0.y = 0.f;
            b1.x = 1.f; b1.y = 0.f;
        }

        int   idx0 = -1, idx1 = -1;
        float hv0 = 0.f, hv1 = 0.f;
#pragma unroll
        for (int g = 0; g < 8; ++g) {
            v8f cz = {};
            // D[M][N]: lane holds pixel N = lane%16; low lanes boxes g*8+0..3
            // (VGPR pairs), high lanes boxes g*8+4..7.
            v8f d0 = __builtin_amdgcn_wmma_f32_16x16x4_f32(
                false, af[g], false, b0, (short)0, cz, false, false);
            v8f d1 = __builtin_amdgcn_wmma_f32_16x16x4_f32(
                false, af[g], false, b1, (short)0, cz, false, false);
            int boxBase = g * 8 + hi4;
#pragma unroll
            for (int k = 0; k < 4; ++k) {
                float wh = sbox[boxBase + k][4];
                float lh = sbox[boxBase + k][5];
                float hv = sbox[boxBase + k][6];
                float lx0 = d0[2 * k], ly0 = d0[2 * k + 1];
                float lx1 = d1[2 * k], ly1 = d1[2 * k + 1];
                // Sequential overwrite: per-lane box indices strictly increase,
                // so the last inside box wins (matches reference semantics).
                if (fabsf(lx0) <= wh && fabsf(ly0) <= lh) { idx0 = boxBase + k; hv0 = hv; }
                if (fabsf(lx1) <= wh && fabsf(ly1) <= lh) { idx1 = boxBase + k; hv1 = hv; }
            }
        }

        // Merge low-box (lanes 0-15) and high-box (lanes 16-31) halves; higher
        // global box index has priority ("later box overwrites earlier").
        int   pi0 = __shfl_xor(idx0, 16, 32);
        float ph0 = __shfl_xor(hv0, 16, 32);
        int   pi1 = __shfl_xor(idx1, 16, 32);
        float ph1 = __shfl_xor(hv1, 16, 32);
        if (pi0 > idx0) { idx0 = pi0; hv0 = ph0; }
        if (pi1 > idx1) { idx1 = pi1; hv1 = ph1; }

        float gt  = hi ? ((idx1 >= 0) ? hv1 : 0.f) : ((idx0 >= 0) ? hv0 : 0.f);
        int   pix = base + lane;   // lane<16 -> tile0 pixel, lane>=16 -> tile1 pixel
        if (pix < NPIX) {
            size_t off = (size_t)b * NPIX + pix;
            float x  = logits[off];
            float hm = hmaps[off];
            bool  pos    = gt > 0.f;                    // gt is 0 or hv>0, so neg == !pos
            float weight = pos ? 5.f : 0.1f;            // POS_W / NEG_W
            float valid  = (pos || hm > 0.f) ? 1.f : 0.f;
            float bce = fmaxf(x, 0.f) - x * gt + log1pf(expf(-fabsf(x)));
            float p   = 1.f / (1.f + expf(-x));
            float pt  = p * gt + (1.f - p) * (1.f - gt);
            float aw  = 0.25f * gt + 0.75f * (1.f - gt);
            float om  = 1.f - pt;
            float fw  = om * om * aw * weight;
            acc_b += weight * bce * valid;
            acc_f += fw * bce * valid;
            acc_c += valid;
        }
    }

    // Wave32 reduction, then cross-wave via LDS.
#pragma unroll
    for (int off = 16; off >= 1; off >>= 1) {
        acc_b += __shfl_xor(acc_b, off, 32);
        acc_f += __shfl_xor(acc_f, off, 32);
        acc_c += __shfl_xor(acc_c, off, 32);
    }
    if (lane == 0) { sred[wave][0] = acc_b; sred[wave][1] = acc_f; sred[wave][2] = acc_c; }
    __syncthreads();
    if (tid == 0) {
        float sb = 0.f, sf = 0.f, sc = 0.f;
        for (int w = 0; w < 8; ++w) { sb += sred[w][0]; sf += sred[w][1]; sc += sred[w][2]; }
        float* out = partial + (size_t)(b * CHUNKS + blockIdx.x) * 4;
        out[0] = sb; out[1] = sf; out[2] = sc; out[3] = 0.f;
    }
}

// Deterministic fixed-order final reduction of all block partials -> scalar.
__global__ void finalize_kernel(const float* __restrict__ partial, float* __restrict__ out)
{
    if (threadIdx.x == 0 && blockIdx.x == 0) {
        float total = 0.f, ns = 0.f;
        for (int b = 0; b < BATCH; ++b) {
            float sb = 0.f, sf = 0.f, sc = 0.f;
            for (int k = 0; k < CHUNKS; ++k) {
                const float* p = partial + (size_t)(b * CHUNKS + k) * 4;
                sb += p[0]; sf += p[1]; sc += p[2];
            }
            float denom = fmaxf(sc, 1.f);
            float comb  = 0.5f * (sb / denom) + 0.5f * (sf / denom);
            if (sc > 0.f) { total += comb; ns += 1.f; }
        }
        out[0] = (ns > 0.f) ? (total / fmaxf(ns, 1.f)) : total;   // LOSS_W = 1
    }
}

extern "C" void kernel_launch(void* const* d_in, const int* in_sizes, int n_in,
                              void* d_out, int out_size, void* d_ws, size_t ws_size,
                              hipStream_t stream)
{
    const float* logits = (const float*)d_in[0];  // attention_logits [4,1,540,540]
    const float* boxes  = (const float*)d_in[1];  // boxes            [4,64,7]
    const float* hmaps  = (const float*)d_in[2];  // height_maps      [4,1,540,540]
    float* partial = (float*)d_ws;                // BATCH*CHUNKS*4 floats = 4 KB

    dim3 grid(CHUNKS, BATCH);
    raster_loss_kernel<<<grid, 256, 0, stream>>>(logits, boxes, hmaps, partial);
    finalize_kernel<<<1, 32, 0, stream>>>(partial, (float*)d_out);
}